// STGCN_16286515986725
// MI455X (gfx1250) — compile-verified
//
#include <hip/hip_runtime.h>

typedef __attribute__((ext_vector_type(2))) float v2f;
typedef __attribute__((ext_vector_type(4))) float v4f;
typedef __attribute__((ext_vector_type(8))) float v8f;

#define F_DIM   262144
#define HID     32
#define NNODES  3
#define TSTEPS  15
#define NROWS   45            // T * N rows of the big GEMM
#define NCOLS   64            // [z | h] gates, 32 cols each (r-gate is dead)
#define KCHUNKS 1024
#define KC      (F_DIM / KCHUNKS)   // 256 K per workgroup
#define KB      64                  // K staged per LDS buffer fill
#define FILLS   (KC / KB)           // 4
#define LDA     68                  // padded LDS strides (bank-conflict break)
#define LDB     68

// ---------------- zero the f32 partial-sum accumulator ----------------
__global__ void zero_acc(float* acc) {
    for (int i = threadIdx.x; i < NROWS * NCOLS; i += blockDim.x) acc[i] = 0.0f;
}

// Stage one KB-deep tile: A row-major [48][KB], B transposed [64 cols][KB].
__device__ __forceinline__ void stage_tile(const float* __restrict__ x,
                                           const float* __restrict__ Wz,
                                           const float* __restrict__ Wh,
                                           int k0, float* bufA, float* bufBt,
                                           int tid) {
    // ---- A: 48 rows x 64 K (rows >= 45 zero padded); 768 b128 granules ----
    #pragma unroll
    for (int it = 0; it < 4; ++it) {
        int idx = tid + it * 192;
        int row = idx >> 4, cv = idx & 15;
        v4f val = {0.f, 0.f, 0.f, 0.f};
        if (row < NROWS)
            val = __builtin_nontemporal_load(
                reinterpret_cast<const v4f*>(x + (size_t)row * F_DIM + k0 + cv * 4));
        *reinterpret_cast<v4f*>(&bufA[row * LDA + cv * 4]) = val;
    }
    // ---- B^T: 64 cols x 64 K; granules of 4 along K: 1024 units ----
    #pragma unroll
    for (int it = 0; it < 6; ++it) {
        int idx = tid + it * 192;
        if (idx < 64 * (KB / 4)) {
            int col = idx & 63, kk4 = (idx >> 6) << 2;
            int gate = col >> 5, c = col & 31;          // gate 0=z, 1=h
            const float* Wsrc = gate ? Wh : Wz;
            v4f v;
            v[0] = __builtin_nontemporal_load(Wsrc + (size_t)(k0 + kk4 + 0) * HID + c);
            v[1] = __builtin_nontemporal_load(Wsrc + (size_t)(k0 + kk4 + 1) * HID + c);
            v[2] = __builtin_nontemporal_load(Wsrc + (size_t)(k0 + kk4 + 2) * HID + c);
            v[3] = __builtin_nontemporal_load(Wsrc + (size_t)(k0 + kk4 + 3) * HID + c);
            *reinterpret_cast<v4f*>(&bufBt[col * LDB + kk4]) = v;
        }
    }
}

// Consume one KB tile: per 8 K -> 3x ds_load_b128, 4x v_wmma_f32_16x16x4_f32.
// Global-k mapping per WMMA: lanes 0-15 (half=0) take k={kk,kk+1}, lanes 16-31
// take k={kk+4,kk+5}; the partner WMMA takes {kk+2,kk+3}/{kk+6,kk+7}. A and B
// use the identical mapping, and summation over K makes the order irrelevant.
__device__ __forceinline__ void compute_tile(const float* bufA, const float* bufBt,
                                             int aBase, int bBase0, int bBase1,
                                             v8f& acc0, v8f& acc1) {
    #pragma unroll
    for (int kk = 0; kk < KB; kk += 8) {
        v4f af  = *reinterpret_cast<const v4f*>(&bufA[aBase + kk]);
        v4f b0f = *reinterpret_cast<const v4f*>(&bufBt[bBase0 + kk]);
        v4f b1f = *reinterpret_cast<const v4f*>(&bufBt[bBase1 + kk]);
        v2f a0  = {af[0],  af[1]},  a1  = {af[2],  af[3]};
        v2f b00 = {b0f[0], b0f[1]}, b01 = {b0f[2], b0f[3]};
        v2f b10 = {b1f[0], b1f[1]}, b11 = {b1f[2], b1f[3]};
        acc0 = __builtin_amdgcn_wmma_f32_16x16x4_f32(false, a0, false, b00,
                                                     (short)0, acc0, false, false);
        acc1 = __builtin_amdgcn_wmma_f32_16x16x4_f32(false, a0, false, b10,
                                                     (short)0, acc1, false, false);
        acc0 = __builtin_amdgcn_wmma_f32_16x16x4_f32(false, a1, false, b01,
                                                     (short)0, acc0, false, false);
        acc1 = __builtin_amdgcn_wmma_f32_16x16x4_f32(false, a1, false, b11,
                                                     (short)0, acc1, false, false);
    }
}

// ---------------- big GEMM: [45, 262144] @ [262144, 64] ----------------
// grid = KCHUNKS blocks, 192 threads (6 waves: m = w%3 -> M tile, g = w/3 -> gate)
__global__ __launch_bounds__(192) void gemm_xw(const float* __restrict__ x,
                                               const float* __restrict__ Wz,
                                               const float* __restrict__ Wh,
                                               float* __restrict__ acc) {
    __shared__ __align__(16) float ldsA[2][48 * LDA];
    __shared__ __align__(16) float ldsBt[2][64 * LDB];

    const int tid  = threadIdx.x;
    const int lane = tid & 31;
    const int w    = tid >> 5;
    const int m    = w % 3;        // M tile (rows m*16 .. m*16+15)
    const int g    = w / 3;        // gate (0=z, 1=h) -> cols g*32 .. g*32+31
    const int l15  = lane & 15;
    const int half = lane >> 4;
    const int kStart = blockIdx.x * KC;

    const int aBase  = (m * 16 + l15) * LDA + 4 * half;
    const int bBase0 = (g * 32 + l15) * LDB + 4 * half;
    const int bBase1 = bBase0 + 16 * LDB;

    v8f acc0 = {};   // cols g*32 + [0,16)
    v8f acc1 = {};   // cols g*32 + [16,32)

    stage_tile(x, Wz, Wh, kStart, ldsA[0], ldsBt[0], tid);
    for (int i = 0; i < FILLS; ++i) {
        __syncthreads();
        if (i + 1 < FILLS)
            stage_tile(x, Wz, Wh, kStart + (i + 1) * KB,
                       ldsA[(i + 1) & 1], ldsBt[(i + 1) & 1], tid);
        compute_tile(ldsA[i & 1], ldsBt[i & 1], aBase, bBase0, bBase1, acc0, acc1);
    }

    // ---- reduce K-split partials: C tile row = v + 8*half, col = lane&15 ----
    #pragma unroll
    for (int v = 0; v < 8; ++v) {
        int r = m * 16 + v + half * 8;
        if (r < NROWS) {
            int c = g * 32 + l15;
            atomicAdd(&acc[r * NCOLS + c],      acc0[v]);
            atomicAdd(&acc[r * NCOLS + c + 16], acc1[v]);
        }
    }
}

// ---------------- tail: GCN norm, gates, relu/transpose/mean, classifier ----------------
// 1 block, 96 threads: thread = (n = tid/32, h = tid%32)
__global__ __launch_bounds__(96) void tgcn_tail(const float* __restrict__ G,   // [45][64]
                                                const float* __restrict__ ew,  // [15][9]
                                                const float* __restrict__ bz,
                                                const float* __restrict__ bh,
                                                const float* __restrict__ Wlz,
                                                const float* __restrict__ blz,
                                                const float* __restrict__ Wlh,
                                                const float* __restrict__ blh,
                                                const float* __restrict__ Wfc,
                                                const float* __restrict__ bfc,
                                                float* __restrict__ out) {
    __shared__ float convZ[NNODES][HID];
    __shared__ float convH[NNODES][HID];
    __shared__ float vec[NNODES * HID];

    const int tid = threadIdx.x;
    const int h   = tid & 31;
    const int n   = tid >> 5;                      // 0..2
    float tacc = 0.f;

    for (int t = 0; t < TSTEPS; ++t) {
        // edge e = src*3 + dst; deg[d] = sum over dst==d
        float wv[9];
        #pragma unroll
        for (int e = 0; e < 9; ++e) wv[e] = ew[t * 9 + e];
        float dinv[3];
        #pragma unroll
        for (int d = 0; d < 3; ++d) {
            float deg = wv[d] + wv[3 + d] + wv[6 + d];
            dinv[d] = deg > 0.f ? rsqrtf(deg) : 0.f;
        }
        // sym-normalized aggregation into (n, h)
        float cz = bz[h], ch = bh[h];
        #pragma unroll
        for (int s = 0; s < 3; ++s) {
            float nrm = dinv[s] * wv[s * 3 + n] * dinv[n];
            const float* row = G + (size_t)(t * 3 + s) * NCOLS;
            cz += nrm * row[h];
            ch += nrm * row[32 + h];
        }
        convZ[n][h] = cz;
        convH[n][h] = ch;
        __syncthreads();

        // Hprev == 0, so only the first 32 rows of Wlz/Wlh matter
        float az = blz[h], ah = blh[h];
        #pragma unroll
        for (int j = 0; j < HID; ++j) {
            az += convZ[n][j] * Wlz[j * HID + h];
            ah += convH[n][j] * Wlh[j * HID + h];
        }
        float Z  = 1.f / (1.f + expf(-az));
        float Ht = tanhf(ah);
        float Hn = (1.f - Z) * Ht;                 // Z*Hprev + (1-Z)*Ht with Hprev=0
        tacc += fmaxf(Hn, 0.f);                    // relu, accumulated over time
        __syncthreads();
    }

    vec[h * NNODES + n] = tacc * (1.f / TSTEPS);   // transpose + flatten + time mean
    __syncthreads();

    if (tid < 2) {
        float y = bfc[tid];
        for (int i = 0; i < NNODES * HID; ++i) y += vec[i] * Wfc[i * 2 + tid];
        out[tid] = y;
    }
}

extern "C" void kernel_launch(void* const* d_in, const int* in_sizes, int n_in,
                              void* d_out, int out_size, void* d_ws, size_t ws_size,
                              hipStream_t stream) {
    const float* x   = (const float*)d_in[0];
    // d_in[1] = edge_index: fixed dense 3x3 graph (src=e/3, dst=e%3), not needed
    const float* ew  = (const float*)d_in[2];
    const float* Wz  = (const float*)d_in[3];
    const float* bz  = (const float*)d_in[4];
    // d_in[5..6] = Wr/br: dead (Hprev==0 => R never affects the output)
    const float* Wh  = (const float*)d_in[7];
    const float* bh  = (const float*)d_in[8];
    const float* Wlz = (const float*)d_in[9];
    const float* blz = (const float*)d_in[10];
    // d_in[11..12] = Wlr/blr: dead
    const float* Wlh = (const float*)d_in[13];
    const float* blh = (const float*)d_in[14];
    const float* Wfc = (const float*)d_in[15];
    const float* bfc = (const float*)d_in[16];

    float* acc = (float*)d_ws;   // [45][64] f32 partial-sum accumulator

    zero_acc<<<1, 256, 0, stream>>>(acc);
    gemm_xw<<<KCHUNKS, 192, 0, stream>>>(x, Wz, Wh, acc);
    tgcn_tail<<<1, 96, 0, stream>>>(acc, ew, bz, bh, Wlz, blz, Wlh, blh, Wfc, bfc,
                                    (float*)d_out);
}